// QLSTM_60327110639964
// MI455X (gfx1250) — compile-verified
//
#include <hip/hip_runtime.h>
#include <math.h>

// ---------------------------------------------------------------------------
// QLSTM for MI455X (gfx1250, wave32).
//
// Math restructuring: the NL=2 trainable RX layers + CNOT ring form a FIXED
// 16x16 unitary per gate.  Encoded state amp[m] = r[m]*(-i)^pop(m) with
// r[m] = prod_w (cos|sin)(y_w/2) real.  Folding the phase into
// V = U*diag((-i)^pop) gives  psi_re = Vre*r, psi_im = Vim*r  (real 16x16
// matvecs -> v_wmma_f32_16x16x4_f32).  Z-expectations + w_out collapse into
// Mw[j][n] = sum_w w_out[j][w]*(-1)^{n_w}  ->  gate = Mw * |psi|^2 + b_out
// (another WMMA).  y's x-part is hoisted out of the scan as a WMMA GEMM.
//
// Workspace layout (floats):
//   [0      .. 262144) YX  = x@w_in[:,8:]^T + b_in   (B*T x 4)
//   [262144 .. +1024)  Vre (WMMA A-operand swizzled, 4 gates x 4 k-slices)
//   [263168 .. +1024)  Vim
//   [264192 .. +256)   Mw  (A-operand swizzled)
//   [264448 .. +512)   WB  (w_in[:,8:] pre-swizzled B-operand for YX GEMM)
// Total ~1.04 MB.
// ---------------------------------------------------------------------------

typedef __attribute__((ext_vector_type(2))) float v2f;
typedef __attribute__((ext_vector_type(8))) float v8f;

#define OFF_YX   0
#define OFF_VARE 262144
#define OFF_VAIM 263168
#define OFF_MW   264192
#define OFF_WB   264448

__device__ __forceinline__ v8f wmma4(v2f a, v2f b, v8f c) {
  // v_wmma_f32_16x16x4_f32 : D = A(16x4) x B(4x16) + C(16x16)
  return __builtin_amdgcn_wmma_f32_16x16x4_f32(false, a, false, b, (short)0, c,
                                               false, false);
}

__device__ __forceinline__ float sigmoid_f(float x) {
  return 1.0f / (1.0f + __expf(-x));
}
__device__ __forceinline__ float tanh_f(float x) {
  float xc = fminf(fmaxf(x, -15.0f), 15.0f);
  float e = __expf(2.0f * xc);
  return (e - 1.0f) / (e + 1.0f);
}

// ---------------------------------------------------------------------------
// Setup: build per-gate folded circuit matrices Vre/Vim (16x16), Mw (8x16),
// and the pre-swizzled w_in B-operand for the YX GEMM.  WMMA 16x16x4-f32
// A-operand layout: element (row M=n, K=4q+p) -> slice q, lane n + 16*(p>=2),
// vreg p&1.  B-operand layout (4x16): slice q, lane j (+16 for K=4q+2..3),
// vreg k&1.  Wire w maps to bit (3-w) of the flat state index.
// ---------------------------------------------------------------------------
__global__ void qlstm_setup(const float* __restrict__ wqf,
                            const float* __restrict__ wqi,
                            const float* __restrict__ wqu,
                            const float* __restrict__ wqo,
                            const float* __restrict__ w_out,
                            const float* __restrict__ w_in,
                            float* __restrict__ ws) {
  int tid = threadIdx.x;
  float* VARE = ws + OFF_VARE;
  float* VAIM = ws + OFF_VAIM;
  float* MW   = ws + OFF_MW;
  float* WB   = ws + OFF_WB;

  if (tid < 64) {                       // one thread per (gate, column m)
    int g = tid >> 4, m = tid & 15;
    const float* wq = (g == 0) ? wqf : (g == 1) ? wqi : (g == 2) ? wqu : wqo;
    float re[16], im[16];
#pragma unroll
    for (int n = 0; n < 16; n++) { re[n] = (n == m) ? 1.0f : 0.0f; im[n] = 0.0f; }

    for (int l = 0; l < 2; l++) {
#pragma unroll
      for (int w = 0; w < 4; w++) {     // RX(W[l,w]) on wire w
        float th = 0.5f * wq[l * 4 + w];
        float cth = cosf(th), sth = sinf(th);
        int mask = 1 << (3 - w);
#pragma unroll
        for (int n = 0; n < 16; n++) {
          if (!(n & mask)) {
            int n1 = n | mask;
            float r0 = re[n], i0 = im[n], r1 = re[n1], i1 = im[n1];
            re[n]  = cth * r0 + sth * i1;  im[n]  = cth * i0 - sth * r1;
            re[n1] = cth * r1 + sth * i0;  im[n1] = cth * i1 - sth * r0;
          }
        }
      }
#pragma unroll
      for (int w = 0; w < 4; w++) {     // CNOT(w, (w+1)%4)
        int cb = 1 << (3 - w);
        int tb = 1 << (3 - ((w + 1) & 3));
#pragma unroll
        for (int n = 0; n < 16; n++) {
          if ((n & cb) && !(n & tb)) {
            int n1 = n | tb;
            float tr = re[n]; re[n] = re[n1]; re[n1] = tr;
            float ti = im[n]; im[n] = im[n1]; im[n1] = ti;
          }
        }
      }
    }

    // fold input phase (-i)^pop(m) into column m, store in A-operand layout
    int pc = __popc(m) & 3;
    int q = m >> 2, p = m & 3;
    int lhalf = (p >= 2) ? 16 : 0;
    int vreg = p & 1;
#pragma unroll
    for (int n = 0; n < 16; n++) {
      float vr, vi;
      if      (pc == 0) { vr =  re[n]; vi =  im[n]; }
      else if (pc == 1) { vr =  im[n]; vi = -re[n]; }
      else if (pc == 2) { vr = -re[n]; vi = -im[n]; }
      else              { vr = -im[n]; vi =  re[n]; }
      int idx = (((g * 4 + q) * 32) + (n + lhalf)) * 2 + vreg;
      VARE[idx] = vr;
      VAIM[idx] = vi;
    }
  } else if (tid < 192) {               // Mw in A-operand layout (rows 8..15 = 0)
    int idx = tid - 64;
    int q = idx >> 5, lane = idx & 31;
    int jj = lane & 15;
    int koff = (lane < 16) ? 0 : 2;
#pragma unroll
    for (int vreg = 0; vreg < 2; vreg++) {
      int n = 4 * q + koff + vreg;
      float val = 0.0f;
      if (jj < 8) {
#pragma unroll
        for (int w = 0; w < 4; w++) {
          float sgn = ((n >> (3 - w)) & 1) ? -1.0f : 1.0f;
          val += w_out[jj * 4 + w] * sgn;
        }
      }
      MW[(q * 32 + lane) * 2 + vreg] = val;
    }
  } else {                              // WB: w_in[:,8:] as B-operand, 8 slices
    for (int e = tid - 192; e < 256; e += 64) {
      int q = e >> 5, lane = e & 31;
      int jj = lane & 15;
      int koff = (lane < 16) ? 0 : 2;
      float v0 = 0.0f, v1 = 0.0f;
      if (jj < 4) {
        v0 = w_in[jj * 40 + 8 + 4 * q + koff];
        v1 = w_in[jj * 40 + 8 + 4 * q + koff + 1];
      }
      WB[(q * 32 + lane) * 2 + 0] = v0;
      WB[(q * 32 + lane) * 2 + 1] = v1;
    }
  }
}

// ---------------------------------------------------------------------------
// Pre-pass WMMA GEMM:  YX[b*T+t][0:4] = x[b][t][:] @ w_in[:,8:40]^T + b_in.
// One wave per 16-row tile; K=32 -> 8 branch-free k-slices of
// v_wmma_f32_16x16x4_f32 (A: global_load_b64 of x, B: pre-swizzled WB).
// ---------------------------------------------------------------------------
__global__ __launch_bounds__(256) void qlstm_yx(const float* __restrict__ x,
                                                const float* __restrict__ b_in,
                                                float* __restrict__ ws) {
  float* YX = ws + OFF_YX;
  const float* WB = ws + OFF_WB;
  int lane = threadIdx.x & 31;
  int wave = blockIdx.x * (blockDim.x >> 5) + (threadIdx.x >> 5);
  int base = wave * 16;                 // 16 rows of B*T = 65536
  int s = lane & 15;
  int koff = (lane < 16) ? 0 : 2;

  v8f acc;
  float bi = b_in[s & 3];               // clamped load + select, no branch
  float cinit = (s < 4) ? bi : 0.0f;
#pragma unroll
  for (int v = 0; v < 8; v++) acc[v] = cinit;

#pragma unroll
  for (int q = 0; q < 8; q++) {
    v2f A  = *(const v2f*)(x + (size_t)(base + s) * 32 + 4 * q + koff);
    v2f Bv = *(const v2f*)(WB + (q * 32 + lane) * 2);
    acc = wmma4(A, Bv, acc);
  }

  if (s < 4) {                          // D(m=v,n=lane) ; rows v / v+8
    int rbase = base + ((lane < 16) ? 0 : 8);
#pragma unroll
    for (int v = 0; v < 8; v++) YX[(size_t)(rbase + v) * 4 + s] = acc[v];
  }
}

// ---------------------------------------------------------------------------
// Scan kernel: grid = 64 blocks (16 samples each), block = 128 = 4 waves.
// Wave g computes gate g (0=f,1=i,2=u,3=o); gates exchanged via LDS
// double-buffer, one barrier per timestep. h,c live in registers (mirrored
// across lane halves so both halves can feed the WMMA B-operand K-split).
// ---------------------------------------------------------------------------
__global__ __launch_bounds__(128) void qlstm_scan(const float* __restrict__ w_in,
                                                  const float* __restrict__ b_out,
                                                  const float* __restrict__ w_fc,
                                                  const float* __restrict__ b_fc,
                                                  const float* __restrict__ ws,
                                                  float* __restrict__ out) {
  int lane = threadIdx.x & 31;
  int g = threadIdx.x >> 5;             // gate owned by this wave
  int s = lane & 15;                    // sample within tile
  int b = blockIdx.x * 16 + s;
  bool lo = lane < 16;

  const float* VARE = ws + OFF_VARE;
  const float* VAIM = ws + OFF_VAIM;
  const float* MW   = ws + OFF_MW;
  const float* yxp  = ws + OFF_YX + (size_t)b * 64 * 4;

  // fixed matrices, pre-swizzled as A-operands
  v2f vre[4], vim[4], mw[4];
#pragma unroll
  for (int q = 0; q < 4; q++) {
    vre[q] = *(const v2f*)(VARE + (((g * 4 + q) * 32) + lane) * 2);
    vim[q] = *(const v2f*)(VAIM + (((g * 4 + q) * 32) + lane) * 2);
    mw[q]  = *(const v2f*)(MW + ((q * 32 + lane) * 2));
  }

  float win[32];                        // w_in[w][j], recurrent part
#pragma unroll
  for (int k = 0; k < 32; k++) win[k] = w_in[(k >> 3) * 40 + (k & 7)];
  float boutv[8];
#pragma unroll
  for (int v = 0; v < 8; v++) boutv[v] = lo ? b_out[v] : 0.0f;

  float h[8], c[8];
#pragma unroll
  for (int v = 0; v < 8; v++) { h[v] = 0.0f; c[v] = 0.0f; }

  __shared__ float gbuf[2][4][16][8];   // [buf][gate][sample][j]
  int buf = 0;

  for (int t = 0; t < 64; t++) {
    // ---- y = YX[b][t] + h @ w_in[:, :8]^T  (mirrored across lane halves)
    float4 yx = *(const float4*)(yxp + t * 4);
    __builtin_prefetch(yxp + (t + 1) * 4, 0, 0);   // overlap next step's load
    float y0 = yx.x, y1 = yx.y, y2 = yx.z, y3 = yx.w;
#pragma unroll
    for (int j = 0; j < 8; j++) {
      y0 += win[j]      * h[j];
      y1 += win[8 + j]  * h[j];
      y2 += win[16 + j] * h[j];
      y3 += win[24 + j] * h[j];
    }

    // ---- encoded real amplitudes r[m] = prod (cos|sin)(y_w/2)
    float c0 = __cosf(0.5f * y0), s0 = __sinf(0.5f * y0);
    float c1 = __cosf(0.5f * y1), s1 = __sinf(0.5f * y1);
    float c2 = __cosf(0.5f * y2), s2 = __sinf(0.5f * y2);
    float c3 = __cosf(0.5f * y3), s3 = __sinf(0.5f * y3);
    float a01[4] = {c0 * c1, c0 * s1, s0 * c1, s0 * s1};
    float a23_0 = c2 * c3, a23_1 = c2 * s3, a23_2 = s2 * c3, a23_3 = s2 * s3;
    float x0 = lo ? a23_0 : a23_2;      // lane-half K-split of B operand
    float x1 = lo ? a23_1 : a23_3;

    // ---- Psi = V * r : 8 WMMAs (re + im, 4 K-slices each)
    v8f aR, aI;
#pragma unroll
    for (int v = 0; v < 8; v++) { aR[v] = 0.0f; aI[v] = 0.0f; }
#pragma unroll
    for (int q = 0; q < 4; q++) {
      v2f Bq; Bq[0] = a01[q] * x0; Bq[1] = a01[q] * x1;  // r[4q+koff(+1)]
      aR = wmma4(vre[q], Bq, aR);
      aI = wmma4(vim[q], Bq, aI);
    }

    // ---- probabilities, gather full state per lane via half-swap
    float P[8], Q[8];
#pragma unroll
    for (int v = 0; v < 8; v++) P[v] = aR[v] * aR[v] + aI[v] * aI[v];
#pragma unroll
    for (int v = 0; v < 8; v++) Q[v] = __shfl_xor(P[v], 16, 32);

    // ---- gate_pre = Mw * P + b_out : build B operand, 4 WMMAs
    v2f Bg0, Bg1, Bg2, Bg3;
    Bg0[0] = lo ? P[0] : Q[2]; Bg0[1] = lo ? P[1] : Q[3];
    Bg1[0] = lo ? P[4] : Q[6]; Bg1[1] = lo ? P[5] : Q[7];
    Bg2[0] = lo ? Q[0] : P[2]; Bg2[1] = lo ? Q[1] : P[3];
    Bg3[0] = lo ? Q[4] : P[6]; Bg3[1] = lo ? Q[5] : P[7];
    v8f aG;
#pragma unroll
    for (int v = 0; v < 8; v++) aG[v] = boutv[v];
    aG = wmma4(mw[0], Bg0, aG);
    aG = wmma4(mw[1], Bg1, aG);
    aG = wmma4(mw[2], Bg2, aG);
    aG = wmma4(mw[3], Bg3, aG);

    // ---- mirror gate values to the upper lane half, activate, publish
    float gv[8];
#pragma unroll
    for (int v = 0; v < 8; v++) {
      float tv = __shfl_xor(aG[v], 16, 32);
      float pre = lo ? aG[v] : tv;
      gv[v] = (g == 2) ? tanh_f(pre) : sigmoid_f(pre);
    }
    if (lo) {
#pragma unroll
      for (int v = 0; v < 8; v++) gbuf[buf][g][s][v] = gv[v];
    }
    __syncthreads();

    // ---- LSTM cell update (redundant in all waves / both halves)
#pragma unroll
    for (int v = 0; v < 8; v++) {
      float F = gbuf[buf][0][s][v];
      float I = gbuf[buf][1][s][v];
      float U = gbuf[buf][2][s][v];
      float O = gbuf[buf][3][s][v];
      c[v] = F * c[v] + I * U;
      h[v] = O * tanh_f(c[v]);
    }
    buf ^= 1;
  }

  if (g == 0 && lo) {                   // out = h @ w_fc^T + b_fc
    float o = b_fc[0];
#pragma unroll
    for (int v = 0; v < 8; v++) o += w_fc[v] * h[v];
    out[b] = o;
  }
}

// ---------------------------------------------------------------------------
extern "C" void kernel_launch(void* const* d_in, const int* in_sizes, int n_in,
                              void* d_out, int out_size, void* d_ws, size_t ws_size,
                              hipStream_t stream) {
  const float* x     = (const float*)d_in[0];   // (1024, 64, 32)
  const float* w_in  = (const float*)d_in[1];   // (4, 40)
  const float* b_in  = (const float*)d_in[2];   // (4,)
  const float* w_out = (const float*)d_in[3];   // (8, 4)
  const float* b_out = (const float*)d_in[4];   // (8,)
  const float* wq_f  = (const float*)d_in[5];   // (2, 4)
  const float* wq_i  = (const float*)d_in[6];
  const float* wq_u  = (const float*)d_in[7];
  const float* wq_o  = (const float*)d_in[8];
  const float* w_fc  = (const float*)d_in[9];   // (1, 8)
  const float* b_fc  = (const float*)d_in[10];  // (1,)
  float* out = (float*)d_out;                   // (1024, 1)
  float* ws  = (float*)d_ws;                    // needs ~1.04 MB

  qlstm_setup<<<1, 256, 0, stream>>>(wq_f, wq_i, wq_u, wq_o, w_out, w_in, ws);
  qlstm_yx<<<512, 256, 0, stream>>>(x, b_in, ws);            // 4096 tiles
  qlstm_scan<<<64, 128, 0, stream>>>(w_in, b_out, w_fc, b_fc, ws, out);
}